// LGCLEncoder_40492951666757
// MI455X (gfx1250) — compile-verified
//
#include <hip/hip_runtime.h>
#include <stdint.h>

// ext-vector types for WMMA fragments (f32 16x16x4: A=16x4 -> 2 VGPR, B=4x16 -> 2 VGPR, C/D=16x16 -> 8 VGPR)
typedef float v2f __attribute__((ext_vector_type(2)));
typedef float v8f __attribute__((ext_vector_type(8)));
typedef unsigned int v4u __attribute__((ext_vector_type(4)));
typedef int v4i __attribute__((ext_vector_type(4)));
typedef int v8i __attribute__((ext_vector_type(8)));

#if defined(__AMDGCN__) && __has_builtin(__builtin_amdgcn_tensor_load_to_lds)
  #if defined(__clang_major__) && (__clang_major__ >= 23)
    #define LGCL_TDM_6ARG 1
  #else
    #define LGCL_TDM_5ARG 1
  #endif
#endif

// Native non-returning f32 global atomic add (avoids any CAS-loop lowering).
__device__ __forceinline__ void atomic_add_f32(float* p, float v) {
    asm volatile("global_atomic_add_f32 %0, %1, off" : : "v"(p), "v"(v) : "memory");
}

// ---------------------------------------------------------------------------
// TDM: load a 2D row-major f32 tile [dim1 x dim0] (stride0 elems) into LDS.
// Issued by the calling wave; waits on tensorcnt itself; caller must barrier.
// D# layout per CDNA5 ISA ch.10.8 (group0: ctrl/lds/global addr, group1:
// dims/strides, groups 2-3 zero for 2D tensors).
// ---------------------------------------------------------------------------
__device__ __forceinline__ void tdm_load_tile_f32(const float* gsrc,
                                                  unsigned lds_byte_off,
                                                  unsigned dim0, unsigned dim1,
                                                  unsigned stride0) {
#if defined(LGCL_TDM_5ARG) || defined(LGCL_TDM_6ARG)
    unsigned long long ga = (unsigned long long)(uintptr_t)gsrc;
    v4u g0;
    g0.x = 1u;                                             // count=1, user desc, no gather
    g0.y = lds_byte_off;                                   // lds_addr (bytes)
    g0.z = (unsigned)(ga & 0xFFFFFFFFu);                   // global_addr[31:0]
    g0.w = (unsigned)((ga >> 32) & 0x01FFFFFFu) | (2u << 30); // addr[56:32] | type=2
    v8i g1;
    g1[0] = (int)(2u << 16);                               // data_size = 4 bytes
    g1[1] = (int)(dim0 << 16);                             // tensor_dim0[15:0] @ bits63:48
    g1[2] = (int)((dim0 >> 16) | (dim1 << 16));            // dim0 hi | tensor_dim1 lo
    g1[3] = (int)((dim1 >> 16) | (dim0 << 16));            // dim1 hi | tile_dim0
    g1[4] = (int)(dim1 & 0xFFFFu);                         // tile_dim1 (tile_dim2=0)
    g1[5] = (int)stride0;                                  // tensor_dim0_stride[31:0]
    g1[6] = 0;                                             // stride0 hi | dim1_stride lo
    g1[7] = 0;
    v4i gz = {0, 0, 0, 0};
#if defined(LGCL_TDM_5ARG)
    __builtin_amdgcn_tensor_load_to_lds(g0, g1, gz, gz, 0);
#else
    v8i gz8 = {0, 0, 0, 0, 0, 0, 0, 0};
    __builtin_amdgcn_tensor_load_to_lds(g0, g1, gz, gz, gz8, 0);
#endif
    __builtin_amdgcn_s_wait_tensorcnt(0);
#endif
}

// ---------------------------------------------------------------------------
// Edge scatter: z_u[row] += val * srcI[col];  z_i[col] += val * srcU[row]
// 8 threads per edge, each handling 8 columns (two float4 loads, 8 atomics/dir).
// Dominant cost of the whole problem: ~13 GB of traffic; z arrays fit in the
// 192 MB L2 so atomics resolve in-cache.
// ---------------------------------------------------------------------------
__global__ void lgcl_scatter(const float* __restrict__ uSrc,   // [U,64]
                             const float* __restrict__ iSrc,   // [I,64]
                             const float* __restrict__ val,
                             const int*   __restrict__ row,
                             const int*   __restrict__ col,
                             float* __restrict__ zu,           // [U,64]
                             float* __restrict__ zi,           // [I,64]
                             int nE) {
    long long t = (long long)blockIdx.x * blockDim.x + threadIdx.x;
    int e    = (int)(t >> 3);
    int part = (int)(t & 7) * 8;          // starting column of this thread's 8-col slice
    if (e >= nE) return;

    int   r = row[e];
    int   c = col[e];
    float v = val[e];

    // item row -> user accumulator
    {
        const float4* ps = (const float4*)(iSrc + (size_t)c * 64 + part);
        float a[8];
        *(float4*)(a)     = ps[0];
        *(float4*)(a + 4) = ps[1];
        float* d = zu + (size_t)r * 64 + part;
#pragma unroll
        for (int q = 0; q < 8; ++q) atomic_add_f32(d + q, v * a[q]);
    }
    // user row -> item accumulator
    {
        const float4* ps = (const float4*)(uSrc + (size_t)r * 64 + part);
        float a[8];
        *(float4*)(a)     = ps[0];
        *(float4*)(a + 4) = ps[1];
        float* d = zi + (size_t)c * 64 + part;
#pragma unroll
        for (int q = 0; q < 8; ++q) atomic_add_f32(d + q, v * a[q]);
    }
}

// ---------------------------------------------------------------------------
// outM[64,64] += svd^T @ (emb + z)   (K = rows, reduced with V_WMMA_F32_16X16X4_F32)
// 128 threads = 4 waves; wave w owns output M-rows [16w, 16w+16), all 64 N cols
// (4 C tiles = 32 accumulator VGPRs). Grid strides over K in steps of 4.
// ---------------------------------------------------------------------------
__global__ void lgcl_proj(const float* __restrict__ svd,   // [rows,64] ; A[m,k] = svd[k,m]
                          const float* __restrict__ emb,   // [rows,64] ; B[k,n]
                          const float* __restrict__ z,     // [rows,64] ; added to B
                          int rows,
                          float* __restrict__ outM) {      // [64,64]
    int lane  = threadIdx.x & 31;
    int mt    = threadIdx.x >> 5;       // M-tile (0..3)
    int lid   = lane & 15;
    int khalf = (lane >> 4) * 2;        // low lanes K+0/K+1, high lanes K+2/K+3
    int m     = mt * 16 + lid;

    v8f c[4] = {};                      // 4 N-tiles of 16x16 f32

    int kstride = (int)gridDim.x * 4;
    for (int k0 = (int)blockIdx.x * 4; k0 < rows; k0 += kstride) {
        int k = k0 + khalf;
        v2f a;
        a.x = svd[(size_t)k * 64 + m];
        a.y = svd[(size_t)(k + 1) * 64 + m];
#pragma unroll
        for (int nt = 0; nt < 4; ++nt) {
            int n = nt * 16 + lid;
            v2f b;
            b.x = emb[(size_t)k * 64 + n]       + z[(size_t)k * 64 + n];
            b.y = emb[(size_t)(k + 1) * 64 + n] + z[(size_t)(k + 1) * 64 + n];
            c[nt] = __builtin_amdgcn_wmma_f32_16x16x4_f32(
                false, a, false, b, (short)0, c[nt], false, false);
        }
    }

    // Fold partial 64x64 result into global accumulator.
    int rbase = (lane >> 4) * 8;        // C layout: VGPR r -> M=r (low lanes) / M=r+8 (high)
#pragma unroll
    for (int nt = 0; nt < 4; ++nt) {
#pragma unroll
        for (int r = 0; r < 8; ++r) {
            int M = mt * 16 + rbase + r;
            int N = nt * 16 + lid;
            atomic_add_f32(&outM[(size_t)M * 64 + N], c[nt][r]);
        }
    }
}

// ---------------------------------------------------------------------------
// E gather: out[b,:] = emb[idx[b],:] + z1[idx[b],:] + z2[idx[b],:]
// ---------------------------------------------------------------------------
__global__ void lgcl_gatherE(const float* __restrict__ emb,
                             const float* __restrict__ z1,
                             const float* __restrict__ z2,
                             const int*   __restrict__ idx,
                             float* __restrict__ out, int batch) {
    int t = blockIdx.x * blockDim.x + threadIdx.x;
    int b = t >> 4;                      // 16 float4 slices per 64-col row
    int j = (t & 15) * 4;
    if (b >= batch) return;
    int g = idx[b];
    float4 e = *(const float4*)(emb + (size_t)g * 64 + j);
    float4 a = *(const float4*)(z1  + (size_t)g * 64 + j);
    float4 c = *(const float4*)(z2  + (size_t)g * 64 + j);
    float4 o;
    o.x = e.x + a.x + c.x;
    o.y = e.y + a.y + c.y;
    o.z = e.z + a.z + c.z;
    o.w = e.w + a.w + c.w;
    *(float4*)(out + (size_t)b * 64 + j) = o;
}

// ---------------------------------------------------------------------------
// G gather: out[b,:] = (svd[idx[b],:] * svd_s) @ Mm   via WMMA (K=64, 16 steps)
// 128 threads = 4 waves; each wave handles one 16-row batch tile.
// Mm (16 KB) is staged into LDS once per block by the Tensor Data Mover, then
// all B fragments are served from LDS (ds_load) instead of re-reading global.
// ---------------------------------------------------------------------------
__global__ void lgcl_gatherG(const float* __restrict__ svd,   // [rows,64]
                             const float* __restrict__ ss,    // [64]
                             const int*   __restrict__ idx,   // [batch]
                             const float* __restrict__ Mm,    // [64,64]
                             float* __restrict__ out, int batch) {
    __shared__ float smem[64 * 64];

    int lane    = threadIdx.x & 31;
    int wave    = threadIdx.x >> 5;

#if defined(LGCL_TDM_5ARG) || defined(LGCL_TDM_6ARG)
    if (wave == 0) {
        unsigned lds_off = (unsigned)(uintptr_t)&smem[0];   // flat LDS addr low 32b = LDS offset
        tdm_load_tile_f32(Mm, lds_off, 64u, 64u, 64u);      // includes s_wait_tensorcnt 0
    }
#else
    for (int i = threadIdx.x; i < 64 * 64 / 4; i += (int)blockDim.x)
        ((float4*)smem)[i] = ((const float4*)Mm)[i];
#endif
    __syncthreads();

    int rowbase = ((int)blockIdx.x * 4 + wave) * 16;
    if (rowbase >= batch) return;

    int lid   = lane & 15;
    int khalf = (lane >> 4) * 2;
    int rr    = rowbase + lid;
    if (rr >= batch) rr = batch - 1;     // clamp (EXEC must stay all-ones for WMMA)
    int g = idx[rr];

    v8f c[4] = {};
#pragma unroll
    for (int k0 = 0; k0 < 64; k0 += 4) {
        int k = k0 + khalf;
        v2f a;
        a.x = svd[(size_t)g * 64 + k]     * ss[k];
        a.y = svd[(size_t)g * 64 + k + 1] * ss[k + 1];
#pragma unroll
        for (int nt = 0; nt < 4; ++nt) {
            int n = nt * 16 + lid;
            v2f b;
            b.x = smem[k * 64 + n];
            b.y = smem[(k + 1) * 64 + n];
            c[nt] = __builtin_amdgcn_wmma_f32_16x16x4_f32(
                false, a, false, b, (short)0, c[nt], false, false);
        }
    }

    int rbase = (lane >> 4) * 8;
#pragma unroll
    for (int nt = 0; nt < 4; ++nt) {
#pragma unroll
        for (int r = 0; r < 8; ++r) {
            int orow = rowbase + rbase + r;
            if (orow < batch)
                out[(size_t)orow * 64 + nt * 16 + lid] = c[nt][r];
        }
    }
}

// ---------------------------------------------------------------------------
extern "C" void kernel_launch(void* const* d_in, const int* in_sizes, int n_in,
                              void* d_out, int out_size, void* d_ws, size_t ws_size,
                              hipStream_t stream) {
    const float* user_emb = (const float*)d_in[0];
    const float* item_emb = (const float*)d_in[1];
    const float* edge_val = (const float*)d_in[2];
    const float* svd_u    = (const float*)d_in[3];
    const float* svd_s    = (const float*)d_in[4];
    const float* svd_v    = (const float*)d_in[5];
    const int*   edge_row = (const int*)d_in[6];
    const int*   edge_col = (const int*)d_in[7];
    const int*   users    = (const int*)d_in[8];
    const int*   items    = (const int*)d_in[9];

    const int U = in_sizes[0] / 64;
    const int I = in_sizes[1] / 64;
    const int E = in_sizes[2];
    const int B = in_sizes[8];

    float* ws = (float*)d_ws;
    const size_t U64 = (size_t)U * 64, I64 = (size_t)I * 64;
    float* zu1  = ws;                       // [U,64] layer-1 user propagation
    float* zi1  = zu1 + U64;                // [I,64]
    float* zu2  = zi1 + I64;                // [U,64] layer-2
    float* zi2  = zu2 + U64;                // [I,64]
    float* Msum = zi2 + I64;                // [64,64] = svd_v^T (item_emb + zi1)
    float* Nsum = Msum + 64 * 64;           // [64,64] = svd_u^T (user_emb + zu1)

    const size_t zero_bytes = (2 * U64 + 2 * I64 + 2 * 64 * 64) * sizeof(float);
    hipMemsetAsync(d_ws, 0, zero_bytes, stream);

    // sparse propagation, layer 1 then layer 2
    long long st = (long long)E * 8;
    int sblocks  = (int)((st + 255) / 256);
    lgcl_scatter<<<sblocks, 256, 0, stream>>>(user_emb, item_emb, edge_val,
                                              edge_row, edge_col, zu1, zi1, E);
    lgcl_scatter<<<sblocks, 256, 0, stream>>>(zu1, zi1, edge_val,
                                              edge_row, edge_col, zu2, zi2, E);

    // 64x64 low-rank projections (WMMA f32 reductions over all items/users)
    lgcl_proj<<<256, 128, 0, stream>>>(svd_v, item_emb, zi1, I, Msum);
    lgcl_proj<<<256, 128, 0, stream>>>(svd_u, user_emb, zu1, U, Nsum);

    float* out = (float*)d_out;
    const size_t sec = (size_t)B * 64;

    // E_u[users], E_i[items]
    int et = B * 16;
    lgcl_gatherE<<<(et + 255) / 256, 256, 0, stream>>>(user_emb, zu1, zu2, users, out,       B);
    lgcl_gatherE<<<(et + 255) / 256, 256, 0, stream>>>(item_emb, zi1, zi2, items, out + sec, B);

    // G_u[users] = (svd_u[users]*s) @ Msum ; G_i[items] = (svd_v[items]*s) @ Nsum
    int gblocks = (B + 63) / 64;
    lgcl_gatherG<<<gblocks, 128, 0, stream>>>(svd_u, svd_s, users, Msum, out + 2 * sec, B);
    lgcl_gatherG<<<gblocks, 128, 0, stream>>>(svd_v, svd_s, items, Nsum, out + 3 * sec, B);
}